// NeuralSort_11845519802856
// MI455X (gfx1250) — compile-verified
//
#include <hip/hip_runtime.h>
#include <math.h>

typedef __attribute__((ext_vector_type(2))) float v2f;
typedef __attribute__((ext_vector_type(8))) float v8f;

#define N    2048
#define NB   16          // j-rows per block tile
#define BLK  256         // 8 wave32 per block
#define K2   1.4426950408889634f   // log2(e) / TAU  (TAU = 1)

// ---------------- Kernel 1: Asum[b,i] = sum_k |s[b,i] - s[b,k]| ----------------
__global__ void neuralsort_asum_kernel(const float* __restrict__ scores,
                                       float* __restrict__ asum) {
    __shared__ float s_sh[N];
    const int tid = threadIdx.x;
    const int b   = blockIdx.y;
    const float* srow = scores + (size_t)b * N;
#pragma unroll
    for (int k = 0; k < N / BLK; ++k)
        s_sh[tid + k * BLK] = srow[tid + k * BLK];
    __syncthreads();

    const int   i  = blockIdx.x * BLK + tid;
    const float si = s_sh[i];
    float acc = 0.0f;
#pragma unroll 8
    for (int k = 0; k < N; ++k)
        acc += fabsf(si - s_sh[k]);          // LDS broadcast read, no conflicts
    asum[(size_t)b * N + i] = acc;
}

// ---------------- Kernel 2: fused P_max + softmax, WMMA max pass ----------------
__global__ void neuralsort_softmax_kernel(const float* __restrict__ scores,
                                          const float* __restrict__ asum,
                                          float* __restrict__ out) {
    __shared__ float s_sh[N];
    __shared__ float a_sh[N];          // Asum * K2  (log2 domain)
    __shared__ float red[NB * 8];      // per-wave partials
    __shared__ float m_sh[NB];
    __shared__ float rinv_sh[NB];

    const int tid  = threadIdx.x;
    const int lane = tid & 31;
    const int w    = tid >> 5;                 // wave id 0..7
    const int b    = blockIdx.y;
    const int j0   = blockIdx.x * NB;

    const float* srow = scores + (size_t)b * N;
    const float* arow = asum   + (size_t)b * N;
#pragma unroll
    for (int k = 0; k < N / BLK; ++k) {
        s_sh[tid + k * BLK] = srow[tid + k * BLK];
        a_sh[tid + k * BLK] = arow[tid + k * BLK] * K2;
    }
    __syncthreads();

    // ---- Pass A: row maxima via V_WMMA_F32_16X16X4_F32 ----
    // D[j,i] = (c_j*K2)*s_i + 1*(-Asum_i*K2)  == t in exp2 domain
    const int  l16 = lane & 15;
    const bool lo  = lane < 16;
    v2f amat;   // A 16x4: lanes 0-15 {K0=c2_j, K1=1}; lanes 16-31 {K2=0, K3=0}
    {
        float cj2 = (float)(N - 1 - 2 * (j0 + l16)) * K2;
        amat.x = lo ? cj2  : 0.0f;
        amat.y = lo ? 1.0f : 0.0f;
    }
    v8f rmax;
#pragma unroll
    for (int r = 0; r < 8; ++r) rmax[r] = -3.402823466e38f;

    // wave w covers 16 of the 128 i-tiles
    for (int t = w * 16; t < w * 16 + 16; ++t) {
        const int ibase = t * 16;
        float sv = s_sh[ibase + l16];
        float av = a_sh[ibase + l16];
        v2f bmat;   // B 4x16: lanes 0-15 {K0=s_i, K1=-a2_i}; lanes 16-31 zero
        bmat.x = lo ?  sv : 0.0f;
        bmat.y = lo ? -av : 0.0f;
        v8f c = {};
        v8f d = __builtin_amdgcn_wmma_f32_16x16x4_f32(
                    /*neg_a=*/false, amat, /*neg_b=*/false, bmat,
                    /*c_mod=*/(short)0, c, /*reuse_a=*/false, /*reuse_b=*/false);
#pragma unroll
        for (int r = 0; r < 8; ++r) rmax[r] = fmaxf(rmax[r], d[r]);
    }
    // cross-lane max inside each 16-lane half (half = distinct row group)
#pragma unroll
    for (int msk = 1; msk <= 8; msk <<= 1) {
#pragma unroll
        for (int r = 0; r < 8; ++r)
            rmax[r] = fmaxf(rmax[r], __shfl_xor(rmax[r], msk, 32));
    }
    if (l16 == 0) {
        const int half = lane >> 4;            // 0 -> rows r, 1 -> rows r+8
#pragma unroll
        for (int r = 0; r < 8; ++r)
            red[(r + half * 8) * 8 + w] = rmax[r];
    }
    __syncthreads();
    if (tid < NB) {
        float m = red[tid * 8];
#pragma unroll
        for (int q = 1; q < 8; ++q) m = fmaxf(m, red[tid * 8 + q]);
        m_sh[tid] = m;
    }
    __syncthreads();

    // ---- Pass B: Z_j = sum_i exp2(t2 - m) ----
    const float c2base = (float)(N - 1 - 2 * j0) * K2;
    float z[NB];
#pragma unroll
    for (int jj = 0; jj < NB; ++jj) z[jj] = 0.0f;
#pragma unroll
    for (int it = 0; it < N / BLK; ++it) {
        const int   i  = tid + it * BLK;
        const float si = s_sh[i];
        const float ai = a_sh[i];
#pragma unroll
        for (int jj = 0; jj < NB; ++jj) {
            float cj2 = c2base - (2.0f * K2) * jj;
            float t2  = fmaf(si, cj2, -ai) - m_sh[jj];
            z[jj] += __builtin_amdgcn_exp2f(t2);
        }
    }
#pragma unroll
    for (int msk = 1; msk <= 16; msk <<= 1) {
#pragma unroll
        for (int jj = 0; jj < NB; ++jj)
            z[jj] += __shfl_xor(z[jj], msk, 32);
    }
    __syncthreads();                  // red[] safe to reuse after this
    if (lane == 0) {
#pragma unroll
        for (int jj = 0; jj < NB; ++jj) red[jj * 8 + w] = z[jj];
    }
    __syncthreads();
    if (tid < NB) {
        float Z = red[tid * 8];
#pragma unroll
        for (int q = 1; q < 8; ++q) Z += red[tid * 8 + q];
        rinv_sh[tid] = 1.0f / Z;
    }
    __syncthreads();

    // ---- Pass C: recompute exp2 and stream out (non-temporal: 268MB >> L2) ----
    const size_t obase = ((size_t)b * N + (size_t)j0) * N;
#pragma unroll
    for (int it = 0; it < N / BLK; ++it) {
        const int   i  = tid + it * BLK;
        const float si = s_sh[i];
        const float ai = a_sh[i];
#pragma unroll
        for (int jj = 0; jj < NB; ++jj) {
            float cj2 = c2base - (2.0f * K2) * jj;
            float t2  = fmaf(si, cj2, -ai) - m_sh[jj];
            float v   = __builtin_amdgcn_exp2f(t2) * rinv_sh[jj];
            __builtin_nontemporal_store(v, &out[obase + (size_t)jj * N + i]);
        }
    }
}

extern "C" void kernel_launch(void* const* d_in, const int* in_sizes, int n_in,
                              void* d_out, int out_size, void* d_ws, size_t ws_size,
                              hipStream_t stream) {
    const float* scores = (const float*)d_in[0];
    float*       out    = (float*)d_out;
    float*       asum   = (float*)d_ws;              // B*N floats = 128 KB scratch
    const int B = in_sizes[0] / N;                   // 16

    neuralsort_asum_kernel<<<dim3(N / BLK, B), dim3(BLK), 0, stream>>>(scores, asum);
    neuralsort_softmax_kernel<<<dim3(N / NB, B), dim3(BLK), 0, stream>>>(scores, asum, out);
}